// DetectionLoss_26371099197476
// MI455X (gfx1250) — compile-verified
//
#include <hip/hip_runtime.h>
#include <cstdint>
#include <math.h>

#define NB    128
#define NANCH 1024
#define NT    64
#define PD    85
#define NLOG  79   // reference: logits = pred[:, 6:] -> 79 columns (col 5 skipped)

typedef float v2f __attribute__((ext_vector_type(2)));
typedef float v8f __attribute__((ext_vector_type(8)));

__device__ __forceinline__ float softplusf(float x) {
  // jax.nn.softplus == logaddexp(x, 0) == max(x,0) + log1p(exp(-|x|))
  return fmaxf(x, 0.0f) + log1pf(expf(-fabsf(x)));
}

__device__ __forceinline__ float smoothl1(float d) {
  float ad = fabsf(d);
  return (ad < 1.0f) ? 0.5f * d * d : ad - 0.5f;
}

__device__ __forceinline__ float blockReduceSum(float v, float* red, int tid) {
  red[tid] = v;
  __syncthreads();
  for (int s = 128; s > 0; s >>= 1) {
    if (tid < s) red[tid] = red[tid] + red[tid + s];
    __syncthreads();
  }
  float r = red[0];
  __syncthreads();
  return r;
}

__device__ __forceinline__ float blockReduceMax(float v, float* red, int tid) {
  red[tid] = v;
  __syncthreads();
  for (int s = 128; s > 0; s >>= 1) {
    if (tid < s) red[tid] = fmaxf(red[tid], red[tid + s]);
    __syncthreads();
  }
  float r = red[0];
  __syncthreads();
  return r;
}

// One block per batch sample. 256 threads = 8 wave32s; 4 anchors per thread.
__global__ __launch_bounds__(256, 2)
void det_loss_per_sample(const float* __restrict__ preds,
                         const float* __restrict__ targets,
                         float* __restrict__ perSample) {
  const int b   = blockIdx.x;
  const int tid = threadIdx.x;
  const float* __restrict__ pred = preds + (size_t)b * NANCH * PD;
  const float* __restrict__ tgt  = targets + (size_t)b * NT * 5;

  __shared__ float sT[NT * 5];
  __shared__ float sArea[NT];
  __shared__ int   sValid[NT];
  __shared__ int   sKv;
  __shared__ float red[256];

  // ---- Stage the 64x5 target table into LDS via CDNA5 async global->LDS ----
  for (int t = tid; t < NT * 5; t += 256) {
    uint32_t lds_off = (uint32_t)(size_t)(&sT[t]);  // flat low 32b == LDS offset
    const float* ga = tgt + t;
    asm volatile("global_load_async_to_lds_b32 %0, %1, off"
                 :: "v"(lds_off), "v"(ga) : "memory");
  }
  asm volatile("s_wait_asynccnt 0" ::: "memory");
  __syncthreads();

  if (tid < NT) {
    const float tx1 = sT[tid * 5 + 0], ty1 = sT[tid * 5 + 1];
    const float tx2 = sT[tid * 5 + 2], ty2 = sT[tid * 5 + 3];
    sArea[tid]  = (tx2 - tx1) * (ty2 - ty1);
    sValid[tid] = (sT[tid * 5 + 4] >= 0.0f) ? 1 : 0;
  }
  __syncthreads();
  if (tid == 0) {
    int c = 0;
    for (int j = 0; j < NT; ++j) c += sValid[j];
    sKv = c;
  }
  __syncthreads();
  const bool hasValid = (sKv > 0);

  // ---- Pass 1: per-anchor best IoU / argmax (first-index tie-break) --------
  float px1[4], py1[4], px2[4], py2[4], pcf[4], bi[4];
  int   bidx[4];
  float lmax = -INFINITY;
  int   lany = 0;
#pragma unroll
  for (int k = 0; k < 4; ++k) {
    const int i = tid + k * 256;
    const float* __restrict__ p = pred + (size_t)i * PD;
    px1[k] = p[0]; py1[k] = p[1]; px2[k] = p[2]; py2[k] = p[3]; pcf[k] = p[4];
    const float a1 = (px2[k] - px1[k]) * (py2[k] - py1[k]);
    float best = -INFINITY;
    int   idx  = 0;
    for (int j = 0; j < NT; ++j) {
      float iou = -1.0f;
      if (sValid[j]) {
        const float xx1 = fmaxf(px1[k], sT[j * 5 + 0]);
        const float yy1 = fmaxf(py1[k], sT[j * 5 + 1]);
        const float xx2 = fminf(px2[k], sT[j * 5 + 2]);
        const float yy2 = fminf(py2[k], sT[j * 5 + 3]);
        const float inter = fmaxf(xx2 - xx1, 0.0f) * fmaxf(yy2 - yy1, 0.0f);
        const float uni   = a1 + sArea[j] - inter;
        iou = inter / (uni + 1e-6f);
      }
      if (iou > best) { best = iou; idx = j; }  // strict > keeps first max
    }
    bi[k] = best; bidx[k] = idx;
    lmax  = fmaxf(lmax, best);
    lany |= (best > 0.5f) ? 1 : 0;
  }

  const float gmax    = blockReduceMax(lmax, red, tid);
  const float anyCnt  = blockReduceSum((float)lany, red, tid);
  const bool  anyMtch = (anyCnt > 0.0f);

  // ---- Pass 2: masked loss accumulation ------------------------------------
  float bboxS = 0.0f, ceS = 0.0f, spmS = 0.0f, spuS = 0.0f, mcntS = 0.0f, spAllS = 0.0f;
#pragma unroll
  for (int k = 0; k < 4; ++k) {
    const int i = tid + k * 256;
    const float* __restrict__ p = pred + (size_t)i * PD;
    float m = 0.0f;
    if (hasValid)
      m = anyMtch ? ((bi[k] > 0.5f) ? 1.0f : 0.0f)
                  : ((bi[k] == gmax) ? 1.0f : 0.0f);
    const int j = bidx[k];
    if (m > 0.0f) {
      bboxS += smoothl1(px1[k] - sT[j * 5 + 0]) + smoothl1(py1[k] - sT[j * 5 + 1])
             + smoothl1(px2[k] - sT[j * 5 + 2]) + smoothl1(py2[k] - sT[j * 5 + 3]);
      // cross-entropy over the 79 logits (pred columns 6..84)
      float mx = -INFINITY;
      for (int c = 0; c < NLOG; ++c) mx = fmaxf(mx, p[6 + c]);
      float se = 0.0f;
      for (int c = 0; c < NLOG; ++c) se += expf(p[6 + c] - mx);
      const float lse = mx + logf(se);
      int lab = (int)sT[j * 5 + 4];                 // trunc toward zero
      lab = lab < 0 ? 0 : (lab > (NLOG - 1) ? (NLOG - 1) : lab);
      ceS  += lse - p[6 + lab];
      spmS += softplusf(-pcf[k]);
      mcntS += 1.0f;
    } else {
      spuS += softplusf(pcf[k]);
    }
    spAllS += softplusf(pcf[k]);
  }

  const float bbox  = blockReduceSum(bboxS,  red, tid);
  const float ce    = blockReduceSum(ceS,    red, tid);
  const float spm   = blockReduceSum(spmS,   red, tid);
  const float spu   = blockReduceSum(spuS,   red, tid);
  const float mcnt  = blockReduceSum(mcntS,  red, tid);
  const float spAll = blockReduceSum(spAllS, red, tid);

  if (tid == 0) {
    const float ucnt      = (float)NANCH - mcnt;
    const float bbox_loss = bbox / fmaxf(mcnt * 4.0f, 1.0f);
    const float cls_loss  = ce   / fmaxf(mcnt, 1.0f);
    const float conf_m    = spm  / fmaxf(mcnt, 1.0f);
    const float conf_u    = spu  / fmaxf(ucnt, 1.0f);
    const float conf_loss = (ucnt > 0.0f) ? 0.5f * (conf_m + conf_u) : conf_m;
    const float loss_valid   = bbox_loss + cls_loss + conf_loss;
    const float loss_novalid = spAll * (1.0f / (float)NANCH);
    perSample[b] = hasValid ? loss_valid : loss_novalid;
  }
}

// Final reduction of the 128 per-sample losses: exact f32 WMMA against a
// ones matrix (D[m][n] = rowsum(A[m])), fixed hardware order => deterministic.
__global__ __launch_bounds__(32)
void det_loss_reduce(const float* __restrict__ perSample, float* __restrict__ out) {
  const int lane = threadIdx.x;  // one full wave32, EXEC all ones
#if __has_builtin(__builtin_amdgcn_wmma_f32_16x16x4_f32)
  // A 16x4 f32 layout: lanes 0-15 -> M=lane, VGPR{0,1}=K{0,1};
  //                    lanes 16-31 -> M=lane-16, VGPR{0,1}=K{2,3}.
  const int m  = lane & 15;
  const int k0 = (lane >> 4) << 1;  // 0 or 2
  v2f a0, a1, bOne;
  a0.x = perSample[m * 4 + k0 + 0];
  a0.y = perSample[m * 4 + k0 + 1];
  a1.x = perSample[64 + m * 4 + k0 + 0];
  a1.y = perSample[64 + m * 4 + k0 + 1];
  bOne.x = 1.0f; bOne.y = 1.0f;     // all-ones B: layout-independent
  v8f acc = {0.f, 0.f, 0.f, 0.f, 0.f, 0.f, 0.f, 0.f};
  acc = __builtin_amdgcn_wmma_f32_16x16x4_f32(false, a0, false, bOne, (short)0, acc, false, false);
  acc = __builtin_amdgcn_wmma_f32_16x16x4_f32(false, a1, false, bOne, (short)0, acc, false, false);
  // D layout: VGPR r holds M=r (lanes 0-15) / M=r+8 (lanes 16-31), N=lane&15.
  // Column 0 sum = (lane0: M=0..7) + (lane16: M=8..15).
  float s = acc[0] + acc[1] + acc[2] + acc[3] + acc[4] + acc[5] + acc[6] + acc[7];
  __shared__ float sh[32];
  sh[lane] = s;
  __syncthreads();
  if (lane == 0) out[0] = (sh[0] + sh[16]) * (1.0f / (float)NB);
#else
  __shared__ float sh[32];
  sh[lane] = perSample[lane] + perSample[lane + 32] + perSample[lane + 64] + perSample[lane + 96];
  __syncthreads();
  if (lane == 0) {
    float t = 0.0f;
    for (int i = 0; i < 32; ++i) t += sh[i];
    out[0] = t * (1.0f / (float)NB);
  }
#endif
}

extern "C" void kernel_launch(void* const* d_in, const int* in_sizes, int n_in,
                              void* d_out, int out_size, void* d_ws, size_t ws_size,
                              hipStream_t stream) {
  const float* preds   = (const float*)d_in[0];   // (128, 1024, 85) f32
  const float* targets = (const float*)d_in[1];   // (128, 64, 5)   f32
  float* ws  = (float*)d_ws;                      // 128 per-sample losses
  float* out = (float*)d_out;                     // scalar

  det_loss_per_sample<<<dim3(NB), dim3(256), 0, stream>>>(preds, targets, ws);
  det_loss_reduce<<<dim3(1), dim3(32), 0, stream>>>(ws, out);
}